// LengthRegulator_31112743092729
// MI455X (gfx1250) — compile-verified
//
#include <hip/hip_runtime.h>
#include <stdint.h>

// ---------------------------------------------------------------------------
// Kernel 1: per-batch inclusive scan of durations (T <= 512) -> cum in d_ws.
// One 512-thread block (16 wave32s) per batch; LDS Hillis-Steele scan.
// ---------------------------------------------------------------------------
__global__ void lr_scan_kernel(const int* __restrict__ dur,
                               int* __restrict__ cum,
                               int T) {
    __shared__ int s[512];
    const int t = threadIdx.x;
    const int b = blockIdx.x;

    s[t] = (t < T) ? dur[(size_t)b * T + t] : 0;
    __syncthreads();

#pragma unroll
    for (int off = 1; off < 512; off <<= 1) {
        int v = (t >= off) ? s[t - off] : 0;
        __syncthreads();
        s[t] += v;
        __syncthreads();
    }

    if (t < T) cum[(size_t)b * T + t] = s[t];
}

// ---------------------------------------------------------------------------
// Kernel 2: gather. Block = (96, 4) = 384 threads (12 wave32s); each block
// owns FRAMES_PER_BLOCK=16 frames of one batch (each thread loops 4 frames).
// cum[b,:] (2 KB) is staged into LDS once per block with the CDNA5 async
// global->LDS b128 path (ASYNCcnt), then each thread binary-searches the LDS
// table (broadcast reads, conflict-free) and moves float4 row segments with
// coalesced global_load_b128 / global_store_b128.
// ---------------------------------------------------------------------------
#define FRAMES_PER_BLOCK 16

__global__ void lr_gather_kernel(const float* __restrict__ x,
                                 const int* __restrict__ cum,
                                 float* __restrict__ out,
                                 int T, int D, int F) {
    __shared__ int s_cum[512];

    const int tid = threadIdx.y * 96 + threadIdx.x;   // 0..383
    const int b   = blockIdx.y;

    // ---- Stage cum[b, 0:T] into LDS via async b128 copy (CDNA5 path) ----
    {
        const uint64_t base = (uint64_t)(uintptr_t)(cum + (size_t)b * T);
        const int nv4 = T >> 2;                       // 128 x 16B
        for (int i = tid; i < nv4; i += 384) {
            // Generic pointers to LDS carry the wave-relative LDS byte
            // offset in their low 32 bits on AMDGCN.
            uint32_t lds_addr = (uint32_t)(uintptr_t)(&s_cum[i * 4]);
            uint32_t goff     = (uint32_t)(i * 16);
            asm volatile("global_load_async_to_lds_b128 %0, %1, %2"
                         :
                         : "v"(lds_addr), "v"(goff), "s"(base)
                         : "memory");
        }
        asm volatile("s_wait_asynccnt 0" ::: "memory");
        __syncthreads();
    }

    const int total = s_cum[T - 1];
    const int nq    = D >> 2;                         // # float4 per row (96)

#pragma unroll
    for (int k = 0; k < FRAMES_PER_BLOCK / 4; ++k) {
        const int f = blockIdx.x * FRAMES_PER_BLOCK + k * 4 + threadIdx.y;
        if (f >= F) continue;

        // Binary search: first idx with cum[idx] > f (searchsorted 'right')
        int lo = 0, hi = T;
        while (lo < hi) {
            int mid = (lo + hi) >> 1;
            if (s_cum[mid] <= f) lo = mid + 1;
            else                 hi = mid;
        }
        int idx = lo < (T - 1) ? lo : (T - 1);        // clamp like reference

        const bool valid = (f < total);
        const float4* __restrict__ src =
            (const float4*)(x + ((size_t)b * T + idx) * (size_t)D);
        float4* __restrict__ dst =
            (float4*)(out + ((size_t)b * F + f) * (size_t)D);

        for (int d4 = threadIdx.x; d4 < nq; d4 += 96) {
            float4 v = make_float4(0.f, 0.f, 0.f, 0.f);
            if (valid) v = src[d4];                   // global_load_b128
            dst[d4] = v;                              // global_store_b128
        }
    }
}

// ---------------------------------------------------------------------------
// Host-side launch. Inputs: d_in[0]=x (f32, B*T*D), d_in[1]=durations (int,
// B*T), d_in[2]=max_len (scalar, static 4096). Dims derived from sizes;
// B=16 per the reference setup.
// ---------------------------------------------------------------------------
extern "C" void kernel_launch(void* const* d_in, const int* in_sizes, int n_in,
                              void* d_out, int out_size, void* d_ws, size_t ws_size,
                              hipStream_t stream) {
    const float* x   = (const float*)d_in[0];
    const int*   dur = (const int*)d_in[1];
    float*       out = (float*)d_out;

    const int B  = 16;
    const int BT = in_sizes[1];          // B*T = 8192
    const int T  = BT / B;               // 512
    const int D  = in_sizes[0] / BT;     // 384
    const int F  = out_size / (B * D);   // 4096 (max_len)

    int* cum = (int*)d_ws;               // B*T ints = 32 KB scratch

    lr_scan_kernel<<<dim3(B), dim3(512), 0, stream>>>(dur, cum, T);

    dim3 blk(96, 4);                     // 384 threads = 12 wave32s
    dim3 grd((F + FRAMES_PER_BLOCK - 1) / FRAMES_PER_BLOCK, B);
    lr_gather_kernel<<<grd, blk, 0, stream>>>(x, cum, out, T, D, F);
}